// ChamferLoss_distance_26259430047858
// MI455X (gfx1250) — compile-verified
//
#include <hip/hip_runtime.h>

typedef __attribute__((ext_vector_type(2))) float v2f;
typedef __attribute__((ext_vector_type(8))) float v8f;

#define NB 16
#define NP 2048
#define TOT (NB * NP)   // 32768 points per input

// Kernel 1: squared norms for both point clouds. i < TOT -> pred, else gt.
__global__ __launch_bounds__(256)
void chamfer_norms(const float* __restrict__ pred,
                   const float* __restrict__ gt,
                   float* __restrict__ n2p,
                   float* __restrict__ n2g) {
  int i = blockIdx.x * 256 + threadIdx.x;          // 0 .. 2*TOT-1
  const float* src = (i < TOT) ? pred : gt;
  float* dst       = (i < TOT) ? n2p  : n2g;
  int j = (i < TOT) ? i : i - TOT;                 // b*NP + n
  int b = j >> 11, n = j & (NP - 1);
  const float* p = src + (size_t)b * 3 * NP + n;
  float x = p[0], y = p[NP], z = p[2 * NP];
  dst[j] = x * x + y * y + z * z;
}

// Kernel 2: one workgroup = (batch b, 16 "column" points). 4 waves sweep all
// 128 "row" tiles. Cross term via V_WMMA_F32_16X16X4_F32 (K=3 padded to 4),
// B pre-scaled by -2 with an exact 0.0 in the K=3 slot — so the A operand's
// K=3 slot may hold any finite value (product is 0) and both A loads are
// unconditional (no EXEC divergence in the hot loop). C is seeded with
// ||a||^2 + ||b||^2 so D = squared distance directly. Running min kept on
// d^2 (min and sqrt commute); one sqrt per output point in the epilogue.
// Run twice with roles swapped to get both chamfer directions.
__global__ __launch_bounds__(128)
void chamfer_pass(const float* __restrict__ Apts,   // row points  [B,3,NP]
                  const float* __restrict__ Bpts,   // col points  [B,3,NP]
                  const float* __restrict__ a2,     // row sq-norms [B*NP]
                  const float* __restrict__ b2,     // col sq-norms [B*NP]
                  float* __restrict__ colmin) {     // [B*NP] min dist per col
  const int b    = blockIdx.x >> 7;
  const int m0   = (blockIdx.x & 127) << 4;
  const int wave = threadIdx.x >> 5;
  const int lane = threadIdx.x & 31;
  const int half = lane >> 4;            // 0: K={0,1}, 1: K={2,3}
  const int l16  = lane & 15;

  const float* Ab  = Apts + (size_t)b * 3 * NP;
  const float* Bb  = Bpts + (size_t)b * 3 * NP;
  const float* a2b = a2 + b * NP;

  const int kb0 = 2 * half;              // 0 or 2 (always a real K plane)
  const int ka1 = half ? 2 : 1;          // clamped: K=3 slot killed by B==0

  // B matrix (4x16, K x N): vgpr j -> K = kb0 + j, N = l16. Scaled by -2.
  // K=3 slot is an exact 0.0 -> zeroes the K=3 contribution of the product.
  v2f bm;
  bm.x = -2.0f * Bb[kb0 * NP + m0 + l16];
  bm.y = half ? 0.0f : -2.0f * Bb[NP + m0 + l16];

  // ||b||^2 for this lane's column (lane-constant across the row sweep).
  const float b2l = b2[b * NP + m0 + l16];

  // Per-lane A base pointers (unconditional loads, simple offset in loop).
  const float* ax_ptr = Ab + kb0 * NP + l16;
  const float* ay_ptr = Ab + ka1 * NP + l16;

  float cmin2 = __builtin_inff();        // running min of squared distance

#pragma unroll 4
  for (int t = wave; t < 128; t += 4) {
    const int n0 = t << 4;
    // A matrix (16x4, M x K): lane -> M = l16, vgpr j -> K slot kb0 + j.
    v2f a;
    a.x = ax_ptr[n0];
    a.y = ay_ptr[n0];                    // upper half: finite dummy (x0 in B)

    // C/D layout: vgpr v, lane L -> M = v + 8*half, N = l16.
    const int nb = n0 + 8 * half;
    v8f c;
#pragma unroll
    for (int v = 0; v < 8; ++v) c[v] = a2b[nb + v] + b2l;

    v8f d2 = __builtin_amdgcn_wmma_f32_16x16x4_f32(
        /*neg_a=*/false, a, /*neg_b=*/false, bm,
        /*c_mod=*/(short)0, c, /*reuse_a=*/false, /*reuse_b=*/false);

#pragma unroll
    for (int v = 0; v < 8; ++v)
      cmin2 = fminf(cmin2, d2[v]);       // 1 VALU op per distance element
  }

  // Reduce the 8 partial column-mins (2 halves x 4 waves) via LDS.
  __shared__ float smem[128];
  smem[wave * 32 + lane] = cmin2;
  __syncthreads();
  if (threadIdx.x < 16) {
    float m = smem[threadIdx.x];
#pragma unroll
    for (int k = 1; k < 8; ++k)
      m = fminf(m, smem[k * 16 + threadIdx.x]);
    // single cheap sqrt per output point (raw v_sqrt_f32)
    colmin[b * NP + m0 + threadIdx.x] =
        __builtin_amdgcn_sqrtf(fmaxf(m, 0.0f));
  }
}

// Kernel 3: deterministic single-block reduction of both min arrays -> loss.
__global__ __launch_bounds__(1024)
void chamfer_reduce(const float* __restrict__ zg,
                    const float* __restrict__ zp,
                    float* __restrict__ out) {
  float s = 0.0f;
  for (int i = threadIdx.x; i < TOT; i += 1024)
    s += zg[i] + zp[i];
  for (int off = 16; off; off >>= 1) s += __shfl_xor(s, off);
  __shared__ float ws[32];
  if ((threadIdx.x & 31) == 0) ws[threadIdx.x >> 5] = s;
  __syncthreads();
  if (threadIdx.x < 32) {
    float t = ws[threadIdx.x];
    for (int off = 16; off; off >>= 1) t += __shfl_xor(t, off);
    if (threadIdx.x == 0) out[0] = t / (float)TOT;
  }
}

extern "C" void kernel_launch(void* const* d_in, const int* in_sizes, int n_in,
                              void* d_out, int out_size, void* d_ws, size_t ws_size,
                              hipStream_t stream) {
  const float* pred = (const float*)d_in[0];
  const float* gt   = (const float*)d_in[1];
  float* out = (float*)d_out;

  float* n2p = (float*)d_ws;        // [TOT] pred squared norms
  float* n2g = n2p + TOT;           // [TOT] gt squared norms
  float* zg  = n2g + TOT;           // [TOT] min dist per gt point   (z)
  float* zp  = zg + TOT;            // [TOT] min dist per pred point (z2)

  chamfer_norms<<<(2 * TOT) / 256, 256, 0, stream>>>(pred, gt, n2p, n2g);
  // z : min over predicted rows for each gt column
  chamfer_pass<<<NB * 128, 128, 0, stream>>>(pred, gt, n2p, n2g, zg);
  // z2: min over gt rows for each predicted column
  chamfer_pass<<<NB * 128, 128, 0, stream>>>(gt, pred, n2g, n2p, zp);
  chamfer_reduce<<<1, 1024, 0, stream>>>(zg, zp, out);
}